// KBSelector_37589553774962
// MI455X (gfx1250) — compile-verified
//
#include <hip/hip_runtime.h>
#include <math.h>

// ---- problem constants (match reference) ----
#define BATCH 4
#define TLEN  1024
#define KCAND 64
#define DKEY  128
#define DCTX  256
#define DVAL  128
#define TTILE 16
#define NTHREADS 256
#define INV_SQRT_DK 0.08838834764831845f   // 1/sqrt(128)

typedef __attribute__((ext_vector_type(2))) float v2f;
typedef __attribute__((ext_vector_type(8))) float v8f;

__global__ __launch_bounds__(NTHREADS)
void kb_selector_kernel(const float* __restrict__ Q,
                        const float* __restrict__ K_kb,
                        const float* __restrict__ V_kb,
                        const float* __restrict__ ctx_vec,
                        const float* __restrict__ W_ctx,
                        const float* __restrict__ b_ctx,
                        const float* __restrict__ rel_emb,
                        const float* __restrict__ ent_emb,
                        const float* __restrict__ q_min,
                        const float* __restrict__ q_max,
                        const float* __restrict__ tau_min,
                        const float* __restrict__ tau_max,
                        const int*   __restrict__ rel_id,
                        const int*   __restrict__ ent_id,
                        float* __restrict__ out_alpha,
                        float* __restrict__ out_v)
{
    __shared__ __align__(16) float sQ [TTILE * DKEY];   // Q tile          (8 KB)
    __shared__ __align__(16) float sQC[TTILE * DCTX];   // q_ctx = Q*W_ctx (16 KB)
    __shared__ __align__(16) float sS [TTILE * KCAND];  // scores -> alpha (4 KB)

    const int tid  = threadIdx.x;
    const int lane = tid & 31;
    const int wave = tid >> 5;
    const int bt0  = blockIdx.x * TTILE;   // flattened (b*T + t) base of this tile

    // ---------- Phase 1: stage Q tile into LDS (coalesced b128) ----------
    for (int i = tid * 4; i < TTILE * DKEY; i += NTHREADS * 4) {
        const float4 qv = *reinterpret_cast<const float4*>(Q + (size_t)bt0 * DKEY + i);
        *reinterpret_cast<float4*>(&sQ[i]) = qv;
    }
    __syncthreads();

    // ---------- Phase 2: q_ctx[t,c] = sum_d Q[t,d] * W_ctx[d,c] ----------
    // 16x128x256 GEMM per tile via V_WMMA_F32_16X16X4_F32 (full fp32).
    // A 16x4 layout: lanes 0-15 hold K=0,1 (M=lane); lanes 16-31 hold K=2,3.
    // B 4x16 layout: mirrored (N=lane&15, K split by lane>>4).
    // D 16x16: VGPR r -> M = r + 8*(lane>>4), N = lane&15.
    {
        const int m16   = lane & 15;
        const int khalf = lane >> 4;          // 0 or 1
        for (int nt = wave * 2; nt < wave * 2 + 2; ++nt) {   // 16 N-tiles over 8 waves
            const int c0 = nt * 16;
            v8f acc = {0.f, 0.f, 0.f, 0.f, 0.f, 0.f, 0.f, 0.f};
            for (int d0 = 0; d0 < DKEY; d0 += 4) {
                const int ka = d0 + khalf * 2;
                v2f a, bb;
                a.x  = sQ[m16 * DKEY + ka];
                a.y  = sQ[m16 * DKEY + ka + 1];
                bb.x = W_ctx[(size_t)(ka    ) * DCTX + c0 + m16];
                bb.y = W_ctx[(size_t)(ka + 1) * DCTX + c0 + m16];
                acc = __builtin_amdgcn_wmma_f32_16x16x4_f32(
                          false, a, false, bb, (short)0, acc, false, false);
            }
#pragma unroll
            for (int r = 0; r < 8; ++r) {
                const int m = r + (khalf << 3);
                sQC[m * DCTX + c0 + m16] = acc[r];
            }
        }
    }
    __syncthreads();

    // ---------- Phase 3: per-(t,k) scores; one wave per (t,k) pair ----------
    // s = inv*(Q . (K_kb + rel + ent + b_ctx)) + inv*(q_ctx . ctx) + IoU
    for (int p = wave; p < TTILE * KCAND; p += 8) {
        const int t  = p >> 6;
        const int k  = p & 63;
        const size_t bt = (size_t)(bt0 + t);
        const size_t rowk = (bt * KCAND + k);
        const float* krow = K_kb    + rowk * DKEY;
        const float* crow = ctx_vec + rowk * DCTX;
        const float* rrow = rel_emb + (size_t)rel_id[rowk] * DKEY;
        const float* erow = ent_emb + (size_t)ent_id[rowk] * DKEY;

        // prefetch next K row this wave will touch (global_prefetch_b8)
        __builtin_prefetch(K_kb + (rowk + 8) * DKEY + lane * 4, 0, 1);

        const int d4 = lane * 4;                 // 128-d dots: 4 floats/lane
        const float4 qv = *reinterpret_cast<const float4*>(&sQ[t * DKEY + d4]);
        const float4 kv = *reinterpret_cast<const float4*>(krow + d4);
        const float4 rv = *reinterpret_cast<const float4*>(rrow + d4);
        const float4 ev = *reinterpret_cast<const float4*>(erow + d4);
        const float4 bv = *reinterpret_cast<const float4*>(b_ctx + d4);

        float acc = qv.x * (kv.x + rv.x + ev.x + bv.x)
                  + qv.y * (kv.y + rv.y + ev.y + bv.y)
                  + qv.z * (kv.z + rv.z + ev.z + bv.z)
                  + qv.w * (kv.w + rv.w + ev.w + bv.w);

        // 256-d context dot: 8 floats/lane as two float4s
        const float4 qc0 = *reinterpret_cast<const float4*>(&sQC[t * DCTX + d4]);
        const float4 qc1 = *reinterpret_cast<const float4*>(&sQC[t * DCTX + 128 + d4]);
        const float4 cv0 = *reinterpret_cast<const float4*>(crow + d4);
        const float4 cv1 = *reinterpret_cast<const float4*>(crow + 128 + d4);
        acc += qc0.x * cv0.x + qc0.y * cv0.y + qc0.z * cv0.z + qc0.w * cv0.w;
        acc += qc1.x * cv1.x + qc1.y * cv1.y + qc1.z * cv1.z + qc1.w * cv1.w;

        acc *= INV_SQRT_DK;

        // wave32 butterfly reduction
#pragma unroll
        for (int off = 16; off > 0; off >>= 1)
            acc += __shfl_xor(acc, off, 32);

        if (lane == 0) {
            const float qmn = q_min[bt], qmx = q_max[bt];
            const float tmn = tau_min[rowk], tmx = tau_max[rowk];
            const float inter = fmaxf(fminf(qmx, tmx) - fmaxf(qmn, tmn), 0.f);
            const float uni   = fmaxf(qmx, tmx) - fminf(qmn, tmn);
            sS[t * KCAND + k] = acc + inter / (uni + 1e-6f);
        }
    }
    __syncthreads();

    // ---------- Phase 4: softmax over k (64) ; 2 rows per wave ----------
    for (int t = wave * 2; t < wave * 2 + 2; ++t) {
        const float s0 = sS[t * KCAND + lane];
        const float s1 = sS[t * KCAND + 32 + lane];
        float mx = fmaxf(s0, s1);
#pragma unroll
        for (int off = 16; off > 0; off >>= 1)
            mx = fmaxf(mx, __shfl_xor(mx, off, 32));
        const float e0 = __expf(s0 - mx);
        const float e1 = __expf(s1 - mx);
        float sm = e0 + e1;
#pragma unroll
        for (int off = 16; off > 0; off >>= 1)
            sm += __shfl_xor(sm, off, 32);
        const float inv = 1.f / sm;
        const float a0 = e0 * inv, a1 = e1 * inv;
        sS[t * KCAND + lane]      = a0;
        sS[t * KCAND + 32 + lane] = a1;
        const size_t bt = (size_t)(bt0 + t);
        out_alpha[bt * KCAND + lane]      = a0;
        out_alpha[bt * KCAND + 32 + lane] = a1;
    }
    __syncthreads();

    // ---------- Phase 5: V_tilde[t,d] = sum_k alpha[t,k] * V_kb[t,k,d] ----------
    // float4 per thread; 32 lanes of a wave share one t -> 512B coalesced rows.
#pragma unroll
    for (int rep = 0; rep < 2; ++rep) {
        const int idx = rep * NTHREADS + tid;        // 0..511
        const int t   = idx >> 5;                    // 0..15
        const int d4  = (idx & 31) * 4;              // 0..124
        const size_t bt = (size_t)(bt0 + t);
        const float* vbase = V_kb + bt * (size_t)(KCAND * DVAL) + d4;
        float ax = 0.f, ay = 0.f, az = 0.f, aw = 0.f;
        for (int k = 0; k < KCAND; ++k) {
            const float al = sS[t * KCAND + k];
            const float4 v = *reinterpret_cast<const float4*>(vbase + k * DVAL);
            ax += al * v.x; ay += al * v.y; az += al * v.z; aw += al * v.w;
        }
        float4 r; r.x = ax; r.y = ay; r.z = az; r.w = aw;
        *reinterpret_cast<float4*>(out_v + bt * DVAL + d4) = r;
    }
}

extern "C" void kernel_launch(void* const* d_in, const int* in_sizes, int n_in,
                              void* d_out, int out_size, void* d_ws, size_t ws_size,
                              hipStream_t stream) {
    const float* Q      = (const float*)d_in[0];
    const float* K_kb   = (const float*)d_in[1];
    const float* V_kb   = (const float*)d_in[2];
    const float* ctxv   = (const float*)d_in[3];
    const float* W_ctx  = (const float*)d_in[4];
    const float* b_ctx  = (const float*)d_in[5];
    const float* rel_e  = (const float*)d_in[6];
    const float* ent_e  = (const float*)d_in[7];
    const float* q_mn   = (const float*)d_in[8];
    const float* q_mx   = (const float*)d_in[9];
    const float* tau_mn = (const float*)d_in[10];
    const float* tau_mx = (const float*)d_in[11];
    const int*   rel_id = (const int*)d_in[12];
    const int*   ent_id = (const int*)d_in[13];

    float* out_alpha = (float*)d_out;                                   // [B,T,K]
    float* out_v     = out_alpha + (size_t)BATCH * TLEN * KCAND;        // [B,T,Dv]

    dim3 grid(BATCH * TLEN / TTILE);   // 256 workgroups
    dim3 block(NTHREADS);              // 8 wave32
    hipLaunchKernelGGL(kb_selector_kernel, grid, block, 0, stream,
                       Q, K_kb, V_kb, ctxv, W_ctx, b_ctx, rel_e, ent_e,
                       q_mn, q_mx, tau_mn, tau_mx, rel_id, ent_id,
                       out_alpha, out_v);
}